// DiffGSTileSampler_18468359372964
// MI455X (gfx1250) — compile-verified
//
#include <hip/hip_runtime.h>
#include <hip/hip_bf16.h>
#include <math.h>

#ifndef __has_builtin
#define __has_builtin(x) 0
#endif

// Problem constants (from reference)
#define P_N   2048
#define W_N   512
#define H_N   512

// Tiling / staging
#define REC   16          // floats per gaussian record (64 B)
#define BATCH 256         // gaussians staged per batch
#define CAP   272         // compacted queue capacity (15 leftover + 256 new)
#define NWAVE 8           // 256 threads / wave32

typedef float v2f __attribute__((ext_vector_type(2)));
typedef float v8f __attribute__((ext_vector_type(8)));
typedef int   v4i __attribute__((ext_vector_type(4)));

#if defined(__gfx1250__) && __has_builtin(__builtin_amdgcn_wmma_f32_16x16x4_f32)
#define USE_WMMA 1
#else
#define USE_WMMA 0
#endif

#if defined(__gfx1250__) && __has_builtin(__builtin_amdgcn_global_load_async_to_lds_b128)
#define USE_ASYNC 1
#else
#define USE_ASYNC 0
#endif

#if defined(__gfx1250__) && __has_builtin(__builtin_amdgcn_exp2f)
#define EXP2F __builtin_amdgcn_exp2f
#else
#define EXP2F exp2f
#endif

// ---------------------------------------------------------------------------
// Async global -> LDS staging helpers (CDNA5 GLOBAL_LOAD_ASYNC_TO_LDS_B128)
// ---------------------------------------------------------------------------
#if USE_ASYNC
typedef __attribute__((address_space(1))) v4i gas_v4i;
typedef __attribute__((address_space(3))) v4i las_v4i;
#endif

__device__ __forceinline__ void async_copy16B(const float* g, float* l) {
#if USE_ASYNC
  __builtin_amdgcn_global_load_async_to_lds_b128(
      (gas_v4i*)(v4i*)(float*)g, (las_v4i*)(v4i*)l, 0, 0);
#else
  *(float4*)l = *(const float4*)g;
#endif
}

template <int N>
__device__ __forceinline__ void wait_async() {
#if USE_ASYNC
#if __has_builtin(__builtin_amdgcn_s_wait_asynccnt)
  __builtin_amdgcn_s_wait_asynccnt(N);
#else
  asm volatile("s_wait_asynccnt %0" ::"i"(N) : "memory");
#endif
#endif
}

// ---------------------------------------------------------------------------
// Prep kernel: bitonic depth sort + per-gaussian precompute into workspace.
// Record layout (16 f32, 64 B):
//   [0:mx 1:my 2:ex 3:ey | 4:ia 5:ib 6:ic 7:- | 8..11:- | 12:op 13:r 14:g 15:b]
// (op/rgb 16B-aligned so the composite loop is one ds_load_b128 per gaussian)
// ---------------------------------------------------------------------------
__global__ __launch_bounds__(1024) void prep_kernel(
    const float* __restrict__ means, const float* __restrict__ covs,
    const float* __restrict__ depth, const float* __restrict__ opac,
    const float* __restrict__ colors, float* __restrict__ ws) {
  __shared__ float key[P_N];
  __shared__ unsigned short sidx[P_N];
  const int tid = threadIdx.x;

  for (int i = tid; i < P_N; i += 1024) {
    key[i] = depth[i];
    sidx[i] = (unsigned short)i;
  }
  for (unsigned k = 2; k <= P_N; k <<= 1) {
    for (unsigned j = k >> 1; j > 0; j >>= 1) {
      __syncthreads();
      for (unsigned i = tid; i < P_N; i += 1024) {
        unsigned ixj = i ^ j;
        if (ixj > i) {
          bool up = ((i & k) == 0);
          float ki = key[i], kj = key[ixj];
          if ((ki > kj) == up) {
            key[i] = kj; key[ixj] = ki;
            unsigned short t = sidx[i]; sidx[i] = sidx[ixj]; sidx[ixj] = t;
          }
        }
      }
    }
  }
  __syncthreads();
  for (int i = tid; i < P_N; i += 1024) {
    int j = sidx[i];
    float a = covs[3 * j], b = covs[3 * j + 1], c = covs[3 * j + 2];
    float det = a * c - b * b;
    det = det > 1e-8f ? det : 1e-8f;
    float ia = c / det, ib = -b / det, ic = a / det;
    float op = opac[j];
    // conservative ellipse half-extents where alpha >= 1/255 is possible
    float dq = ia * ic - ib * ib;   // = 1/det when clamp inactive
    float lv = 255.0f * op;
    float ex = -1e30f, ey = -1e30f;
    if (lv > 1.0f && dq > 0.0f) {
      float t2 = 2.0f * logf(lv);
      ex = sqrtf(t2 * ic / dq) + 1.0f;  // +1px safety margin
      ey = sqrtf(t2 * ia / dq) + 1.0f;
    }
    float* r = ws + (size_t)i * REC;
    r[0] = means[2 * j];  r[1] = means[2 * j + 1];
    r[2] = ex;            r[3] = ey;
    r[4] = ia;  r[5] = ib;  r[6] = ic;  r[7] = 0.0f;
    r[8] = r[9] = r[10] = r[11] = 0.0f;
    r[12] = op;
    r[13] = colors[3 * j]; r[14] = colors[3 * j + 1]; r[15] = colors[3 * j + 2];
  }
}

// ---------------------------------------------------------------------------
// Composite one chunk of 16 (compacted, depth-ordered) gaussians for this
// wave's 16 pixels. Powers via 2x V_WMMA_F32_16X16X4_F32; half-waves
// composite 8 gaussians each, then merge associatively via shfl_xor(16).
// GUARDED=false is the hot path (chunk known full): no limit checks, LDS
// loads become unconditional ds_load_b128 with immediate offsets.
// ---------------------------------------------------------------------------
template <bool GUARDED>
__device__ __forceinline__ void process_chunk(
    const float* comp_s, int cbase, int limit, float cxt, float cyt, bool hi,
    int lane, float xl, float yl, v2f B1, v2f B2, float& T, float& CR,
    float& CG, float& CB) {
  // Build A (coeff) operand: lane L -> gaussian cbase + (L&15)
  int mA = cbase + (lane & 15);
  int rA = GUARDED ? ((mA < limit) ? mA : cbase) : mA;
  const float* r = comp_s + rA * REC;
  float mx = r[0] - cxt;
  float my = r[1] - cyt;
  float ia = r[4], ib = r[5], ic = r[6];
  float c0 = -0.5f * ia * mx * mx - ib * mx * my - 0.5f * ic * my * my;
  float cx = ia * mx + ib * my;
  float cy = ib * mx + ic * my;

  v8f d = {0.f, 0.f, 0.f, 0.f, 0.f, 0.f, 0.f, 0.f};
#if USE_WMMA
  v2f A1, A2;
  A1.x = hi ? cy : c0;            // K = 0 / 2
  A1.y = hi ? (-0.5f * ia) : cx;  // K = 1 / 3
  A2.x = hi ? 0.0f : (-ib);
  A2.y = hi ? 0.0f : (-0.5f * ic);
  d = __builtin_amdgcn_wmma_f32_16x16x4_f32(false, A1, false, B1, (short)0, d,
                                            false, false);
  d = __builtin_amdgcn_wmma_f32_16x16x4_f32(false, A2, false, B2, (short)0, d,
                                            false, false);
#endif

  const float* qbase = comp_s + (cbase + (hi ? 8 : 0)) * REC;
  float Tl = 1.0f, Rl = 0.0f, Gl = 0.0f, Bl = 0.0f;
#pragma unroll
  for (int j = 0; j < 8; ++j) {
    const float* q = qbase + j * REC;
    bool valid = true;
    if (GUARDED) {
      int g = cbase + j + (hi ? 8 : 0);
      valid = (g < limit);
      q = comp_s + ((valid ? g : cbase) * REC);
    }
    float4 orgb = *(const float4*)(q + 12);  // [op, r, g, b], one b128
    float power;
#if USE_WMMA
    power = d[j];
#else
    float dx = xl - (q[0] - cxt), dy = yl - (q[1] - cyt);
    power = -0.5f * (q[4] * dx * dx + q[6] * dy * dy) - q[5] * dx * dy;
#endif
    float op = GUARDED ? (valid ? orgb.x : 0.0f) : orgb.x;
    float e = EXP2F(fminf(power, 0.0f) * 1.4426950408889634f);
    float al = fminf(0.99f, op * e);
    al = ((power > 0.0f) || (al < 0.003921568627451f)) ? 0.0f : al;
    float wgt = al * Tl;
    Rl += orgb.y * wgt;
    Gl += orgb.z * wgt;
    Bl += orgb.w * wgt;
    Tl *= 1.0f - al;
  }
  // merge half-waves in depth order (front = low half)
  float oT = __shfl_xor(Tl, 16);
  float oR = __shfl_xor(Rl, 16);
  float oG = __shfl_xor(Gl, 16);
  float oB = __shfl_xor(Bl, 16);
  float fR = hi ? (oR + oT * Rl) : (Rl + Tl * oR);
  float fG = hi ? (oG + oT * Gl) : (Gl + Tl * oG);
  float fB = hi ? (oB + oT * Bl) : (Bl + Tl * oB);
  float fT = Tl * oT;
  CR += T * fR;
  CG += T * fG;
  CB += T * fB;
  T *= fT;
}

// ---------------------------------------------------------------------------
// Raster kernel: 16x8 pixel tile per workgroup (8 waves, 16 px/wave).
// Batches of 256 sorted gaussians staged async into LDS (double buffered),
// culled vs the tile bbox, compacted order-preserving, composited in WMMA
// chunks of 16.
// ---------------------------------------------------------------------------
__global__ __launch_bounds__(256) void raster_kernel(
    const float* __restrict__ ws, float* __restrict__ out) {
  __shared__ float stage[2][BATCH * REC];
  __shared__ float comp_s[CAP * REC];
  __shared__ unsigned counts[NWAVE];
  __shared__ unsigned s_ncomp;

  const int tid = threadIdx.x;
  const int lane = tid & 31;
  const int wv = tid >> 5;
  const bool hi = (lane >= 16);
  const int px = lane & 15;

  const int tx0 = blockIdx.x * 16;
  const int ty0 = blockIdx.y * 8;
  const float cxt = tx0 + 8.0f;   // tile-local origin (precision)
  const float cyt = ty0 + 4.0f;
  const float xl = (float)px - 7.5f;  // this lane's pixel, local coords
  const float yl = (float)wv - 3.5f;  // this wave's row, local coords

  // B operands (basis): vgpr0 = rows K=0/2, vgpr1 = rows K=1/3
  v2f B1, B2;
  B1.x = hi ? yl : 1.0f;
  B1.y = hi ? (xl * xl) : xl;
  B2.x = hi ? 0.0f : (xl * yl);
  B2.y = hi ? 0.0f : (yl * yl);

  if (tid == 0) s_ncomp = 0;

  // prologue: stage batch 0
  {
    const float* src = ws + (size_t)tid * REC;
    float* dst = &stage[0][tid * REC];
    async_copy16B(src + 0, dst + 0);
    async_copy16B(src + 4, dst + 4);
    async_copy16B(src + 8, dst + 8);
    async_copy16B(src + 12, dst + 12);
  }

  float T = 1.0f, CR = 0.0f, CG = 0.0f, CB = 0.0f;
  const float txmin = tx0 + 0.5f, txmax = tx0 + 15.5f;
  const float tymin = ty0 + 0.5f, tymax = ty0 + 7.5f;

  const int NB = P_N / BATCH;
  for (int b = 0; b < NB; ++b) {
    if (b + 1 < NB) {  // kick next batch before waiting on current
      const float* src = ws + ((size_t)(b + 1) * BATCH + tid) * REC;
      float* dst = &stage[(b + 1) & 1][tid * REC];
      async_copy16B(src + 0, dst + 0);
      async_copy16B(src + 4, dst + 4);
      async_copy16B(src + 8, dst + 8);
      async_copy16B(src + 12, dst + 12);
      wait_async<4>();  // current batch done, next still in flight
    } else {
      wait_async<0>();
    }
    __syncthreads();  // A: staged batch + s_ncomp visible

    // --- cull + order-preserving compaction ---
    const float* rec = &stage[b & 1][tid * REC];
    float4 bb = *(const float4*)rec;  // [mx, my, ex, ey], one b128
    bool keep = (bb.x + bb.z >= txmin) && (bb.x - bb.z <= txmax) &&
                (bb.y + bb.w >= tymin) && (bb.y - bb.w <= tymax);
    unsigned mask = (unsigned)__ballot(keep);
    unsigned prefix = __popc(mask & ((1u << lane) - 1u));
    if (lane == 0) counts[wv] = __popc(mask);
    unsigned ncomp0 = s_ncomp;
    __syncthreads();  // B: counts visible
    unsigned base = ncomp0, total = ncomp0;
    for (int w2 = 0; w2 < NWAVE; ++w2) {
      unsigned cw = counts[w2];
      if (w2 < wv) base += cw;
      total += cw;
    }
    if (keep) {
      const float4* s4 = (const float4*)rec;
      float4* d4 = (float4*)&comp_s[(base + prefix) * REC];
      d4[0] = s4[0];        // mx,my,ex,ey
      d4[1] = s4[1];        // ia,ib,ic,-
      d4[3] = s4[3];        // op,r,g,b
    }
    __syncthreads();  // C: compacted records visible

    // --- composite full chunks of 16 (hot, unguarded path) ---
    unsigned nfull = total & ~15u;
    for (unsigned c = 0; c < nfull; c += 16)
      process_chunk<false>(comp_s, (int)c, (int)total, cxt, cyt, hi, lane, xl,
                           yl, B1, B2, T, CR, CG, CB);
    __syncthreads();  // D: chunk reads done before moving leftovers

    unsigned rem = total - nfull;
    if (nfull > 0 && tid < rem * REC) comp_s[tid] = comp_s[nfull * REC + tid];
    if (tid == 0) s_ncomp = rem;
    // visibility handled by barrier A of next iteration / final barrier
  }
  __syncthreads();
  unsigned remf = s_ncomp;  // flush trailing (<16) gaussians, padded
  for (unsigned c = 0; c < remf; c += 16)
    process_chunk<true>(comp_s, (int)c, (int)remf, cxt, cyt, hi, lane, xl, yl,
                        B1, B2, T, CR, CG, CB);

  if (lane < 16) {
    int x = tx0 + px;
    int y = ty0 + wv;
    int idx = y * W_N + x;
    out[idx] = CR;
    out[W_N * H_N + idx] = CG;
    out[2 * W_N * H_N + idx] = CB;
  }
}

// ---------------------------------------------------------------------------
extern "C" void kernel_launch(void* const* d_in, const int* in_sizes, int n_in,
                              void* d_out, int out_size, void* d_ws,
                              size_t ws_size, hipStream_t stream) {
  (void)in_sizes; (void)n_in; (void)out_size; (void)ws_size;
  const float* means  = (const float*)d_in[0];
  const float* covs   = (const float*)d_in[1];
  const float* depth  = (const float*)d_in[2];
  const float* opac   = (const float*)d_in[3];
  const float* colors = (const float*)d_in[4];
  float* ws  = (float*)d_ws;
  float* out = (float*)d_out;

  prep_kernel<<<1, 1024, 0, stream>>>(means, covs, depth, opac, colors, ws);
  raster_kernel<<<dim3(W_N / 16, H_N / 8), 256, 0, stream>>>(ws, out);
}